// AffinityPropagate_86646670229804
// MI455X (gfx1250) — compile-verified
//
#include <hip/hip_runtime.h>
#include <stdint.h>

// Problem dims (fixed by the reference)
#define BQ 8
#define HQ 352
#define WQ 1216
#define NQ (BQ * HQ * WQ)

#define PROP_TIME 24

// Stencil tile: 64 x 4 pixels per block (64 | 1216, 4 | 352 -> exact tiling)
#define TX 64
#define TY 4
#define HALO_H (TY + 2)       // 6 rows
// LDS tile column c maps to global x = x0 + c - 4.
//   interior span  x in [x0, x0+64)  -> c in [4, 68), 16B-aligned start
//   halo cells     x0-1 -> c=3,  x0+64 -> c=68
#define PITCH 72              // floats per LDS row (72*4 = 288 B, 16B-aligned)

// ---------------------------------------------------------------------------
// CDNA5 async global->LDS copies (ASYNCcnt-tracked)
// ---------------------------------------------------------------------------
__device__ __forceinline__ void async_g2l_b32(const float* gptr, unsigned lds_byte_off) {
    asm volatile("global_load_async_to_lds_b32 %0, %1, off"
                 :
                 : "v"(lds_byte_off), "v"((unsigned long long)(uintptr_t)gptr)
                 : "memory");
}

__device__ __forceinline__ void async_g2l_b128(const float* gptr, unsigned lds_byte_off) {
    asm volatile("global_load_async_to_lds_b128 %0, %1, off"
                 :
                 : "v"(lds_byte_off), "v"((unsigned long long)(uintptr_t)gptr)
                 : "memory");
}

__device__ __forceinline__ void wait_async0() {
#if __has_builtin(__builtin_amdgcn_s_wait_asynccnt)
    __builtin_amdgcn_s_wait_asynccnt(0);
#else
    asm volatile("s_wait_asynccnt 0" ::: "memory");
#endif
}

// ---------------------------------------------------------------------------
// Precompute (runs once):
//   gn[p][k]  = (1 - mask(p)) * |guidance_k|(p + off_k) / wsum(p)   (interleaved x8)
//   base[p]   = mask(p) * blur(p)
//   d0[p]     = blur(p)
// offsets k: (1,1),(1,0),(1,-1),(0,1),(0,-1),(-1,1),(-1,0),(-1,-1)
// Grid (WQ/64, HQ, BQ) x block 64 -> no div/mod address math.
// ---------------------------------------------------------------------------
__global__ __launch_bounds__(64)
void affprop_precompute(const float* __restrict__ guidance,
                        const float* __restrict__ blur,
                        const float* __restrict__ sparse,
                        float* __restrict__ gn,
                        float* __restrict__ base,
                        float* __restrict__ d0) {
    const int x = blockIdx.x * 64 + threadIdx.x;
    const int y = blockIdx.y;
    const int b = blockIdx.z;
    const size_t p = ((size_t)b * HQ + y) * WQ + x;

    const int dy[8] = { 1, 1, 1, 0, 0, -1, -1, -1 };
    const int dx[8] = { 1, 0, -1, 1, -1, 1, 0, -1 };

    float gs[8];
    float wsum = 0.0f;
#pragma unroll
    for (int k = 0; k < 8; ++k) {
        int sy = y + dy[k];
        int sx = x + dx[k];
        float v = 0.0f;
        if (sy >= 0 && sy < HQ && sx >= 0 && sx < WQ) {
            v = fabsf(guidance[((size_t)(b * 8 + k) * HQ + sy) * WQ + sx]);
        }
        gs[k] = v;
        wsum += v;
    }

    float bl = blur[p];
    float mk = (sparse[p] > 0.0f) ? 1.0f : 0.0f;
    float scale = (1.0f - mk) / wsum;   // folds mask-blend + division into the gates

    float4 lo = make_float4(gs[0] * scale, gs[1] * scale, gs[2] * scale, gs[3] * scale);
    float4 hi = make_float4(gs[4] * scale, gs[5] * scale, gs[6] * scale, gs[7] * scale);

    float4* g4 = (float4*)gn;
    g4[p * 2 + 0] = lo;
    g4[p * 2 + 1] = hi;

    base[p] = mk * bl;
    d0[p]   = bl;
}

// ---------------------------------------------------------------------------
// One propagation sweep:  d' = base + sum_k gn_k * d(p + off_k)
// Depth halo staged into LDS: 16x async B128 + 2x async B32 per halo row.
// ---------------------------------------------------------------------------
__global__ __launch_bounds__(TX * TY)
void affprop_step(const float* __restrict__ dsrc,
                  const float* __restrict__ gn,
                  const float* __restrict__ base,
                  float* __restrict__ ddst) {
    __shared__ float tile[HALO_H * PITCH];

    const int tx = threadIdx.x;          // 0..63
    const int ty = threadIdx.y;          // 0..3
    const int x0 = blockIdx.x * TX;
    const int y0 = blockIdx.y * TY;
    const int b  = blockIdx.z;

    const float* dplane = dsrc + (size_t)b * HQ * WQ;

    // ---- Stage halo: 6 rows x (16 B128 interior + left/right B32 halo) = 108 tasks
    const int flat = ty * TX + tx;       // 0..255
    if (flat < HALO_H * 18) {
        const int r  = flat / 18;        // halo row 0..5
        const int s  = flat % 18;        // 0..15: B128 segment, 16: left, 17: right
        const int gy = y0 + r - 1;
        const bool row_ok = (gy >= 0) && (gy < HQ);
        const float* grow = dplane + (size_t)gy * WQ;
        float* lrow = &tile[r * PITCH];

        if (s < 16) {
            // interior: global x = x0 + 4*s (always in [0, WQ)); LDS col 4 + 4*s
            const int c = 4 + 4 * s;
            if (row_ok) {
                async_g2l_b128(grow + (x0 + 4 * s),
                               (unsigned)(uintptr_t)&lrow[c]);
            } else {
                *(float4*)&lrow[c] = make_float4(0.f, 0.f, 0.f, 0.f);
            }
        } else if (s == 16) {
            // left halo: global x = x0 - 1; LDS col 3
            const int gx = x0 - 1;
            if (row_ok && gx >= 0) {
                async_g2l_b32(grow + gx, (unsigned)(uintptr_t)&lrow[3]);
            } else {
                lrow[3] = 0.0f;
            }
        } else {
            // right halo: global x = x0 + TX; LDS col 68
            const int gx = x0 + TX;
            if (row_ok && gx < WQ) {
                async_g2l_b32(grow + gx, (unsigned)(uintptr_t)&lrow[4 + TX]);
            } else {
                lrow[4 + TX] = 0.0f;
            }
        }
    }
    wait_async0();        // each wave drains its own ASYNCcnt ...
    __syncthreads();      // ... then all waves' LDS writes become visible

    // ---- Compute (exact tiling: no bounds checks)
    const size_t p = ((size_t)b * HQ + (y0 + ty)) * WQ + (x0 + tx);

    const float4* g4 = (const float4*)gn + p * 2;
    float4 glo = g4[0];
    float4 ghi = g4[1];

    // center pixel at LDS col tx+4; rows ty..ty+2 (gy-1, gy, gy+1)
    const float* trow0 = &tile[(ty + 0) * PITCH + tx + 3];  // dy = -1
    const float* trow1 = &tile[(ty + 1) * PITCH + tx + 3];  // dy =  0
    const float* trow2 = &tile[(ty + 2) * PITCH + tx + 3];  // dy = +1

    float acc = base[p];
    acc = fmaf(glo.x, trow2[2], acc);   // ( 1,  1)
    acc = fmaf(glo.y, trow2[1], acc);   // ( 1,  0)
    acc = fmaf(glo.z, trow2[0], acc);   // ( 1, -1)
    acc = fmaf(glo.w, trow1[2], acc);   // ( 0,  1)
    acc = fmaf(ghi.x, trow1[0], acc);   // ( 0, -1)
    acc = fmaf(ghi.y, trow0[2], acc);   // (-1,  1)
    acc = fmaf(ghi.z, trow0[1], acc);   // (-1,  0)
    acc = fmaf(ghi.w, trow0[0], acc);   // (-1, -1)

    ddst[p] = acc;
}

// ---------------------------------------------------------------------------
// Launch: precompute once, then 24 ping-pong sweeps; final result in d_out.
// Workspace: gn (N*8 f32) | base (N f32) | dA (N f32)  ~= 137 MB  (< 192 MB L2)
// ---------------------------------------------------------------------------
extern "C" void kernel_launch(void* const* d_in, const int* in_sizes, int n_in,
                              void* d_out, int out_size, void* d_ws, size_t ws_size,
                              hipStream_t stream) {
    const float* guidance = (const float*)d_in[0];  // [B,8,H,W]
    const float* blur     = (const float*)d_in[1];  // [B,1,H,W]
    const float* sparse   = (const float*)d_in[2];  // [B,1,H,W]
    float* out = (float*)d_out;                     // [B,1,H,W]

    float* gn   = (float*)d_ws;                     // N*8 floats (interleaved gates)
    float* base = gn + (size_t)NQ * 8;              // N floats
    float* dA   = base + (size_t)NQ;                // N floats (ping-pong partner)

    (void)in_sizes; (void)n_in; (void)out_size; (void)ws_size;

    dim3 pgrid(WQ / 64, HQ, BQ);
    affprop_precompute<<<pgrid, 64, 0, stream>>>(guidance, blur, sparse, gn, base, out);

    dim3 grid(WQ / TX, HQ / TY, BQ);   // 19 x 88 x 8
    dim3 block(TX, TY, 1);

    const float* src = out;  // d0 = blur lives in d_out
    float*       dst = dA;
    for (int it = 0; it < PROP_TIME; ++it) {
        affprop_step<<<grid, block, 0, stream>>>(src, gn, base, dst);
        float* ns = dst;
        dst = (float*)src;
        src = ns;
    }
    // PROP_TIME = 24 (even): the final sweep wrote d_out.
}